// Attention_18038862643252
// MI455X (gfx1250) — compile-verified
//
#include <hip/hip_runtime.h>

// CDNA5 / gfx1250 fused cross-attention (flash-attention style), fp32 WMMA,
// async global->LDS staging (ASYNCcnt) with double-buffered K/V blocks and
// double-buffered score-reduction scratch (one barrier per key block).
//
// h_enc: [B=16, T_enc=2048, D=1024] f32   (K and V)
// h_dec: [T_dec=512, B=16, D=1024] f32    (Q)
// out  : [B=16, T_dec=512, D=1024] f32
//
// Per workgroup: one batch b, one 16-query tile. 8 waves split D into 128-wide
// chunks. Online softmax over the 2048 keys in blocks of 16.

typedef float v2f __attribute__((ext_vector_type(2)));
typedef float v8f __attribute__((ext_vector_type(8)));

#define TENC   2048
#define TDEC   512
#define DIM    1024
#define NBATCH 16
#define WAVES  8
#define DCHUNK 128          // per-wave slice of D
#define NKEYS  16           // key block size
#define KROW   132          // padded LDS row stride in floats (2*64+4 -> bank-friendly)
#define NKB    (TENC / NKEYS)
#define SREDSZ (8 * WAVES * 32)   // one score-reduce buffer, floats

__global__ __launch_bounds__(256)
void xattn_flash_f32(const float* __restrict__ h_enc,
                     const float* __restrict__ h_dec,
                     float* __restrict__ out)
{
    extern __shared__ float smem[];

    const int tid  = threadIdx.x;
    const int wave = tid >> 5;       // 0..7
    const int lane = tid & 31;       // wave32
    const int half = lane >> 4;      // 0: lanes 0-15, 1: lanes 16-31
    const int lm   = lane & 15;

    const int b  = blockIdx.y;       // batch
    const int q0 = blockIdx.x * 16;  // query-tile base (t_dec)

    // LDS partition:
    //   [2][WAVES][16][KROW]  double-buffered per-wave K/V chunks
    //   [2][8][WAVES][32]     double-buffered score-reduce scratch (i, wave, lane)
    //   [WAVES][16][17]       per-wave P redistribution scratch
    float* sred0 = smem + 2 * WAVES * NKEYS * KROW;
    float* pbuf  = sred0 + 2 * SREDSZ + wave * (16 * 17);

    const int d0 = wave * DCHUNK;
    const float* encb = h_enc + (size_t)b * TENC * DIM;

    // ---------- async staging: this wave's 16x128 chunk of key block kb2 ----------
    auto issue_kblock = [&](int kb2) {
        float* dst = smem + (((kb2 & 1) * WAVES + wave) * (NKEYS * KROW));
        const float* src = encb + (size_t)(kb2 * NKEYS) * DIM + d0 + lane * 4;
#pragma unroll
        for (int r = 0; r < NKEYS; ++r) {
            unsigned lds_off = (unsigned)(uintptr_t)(dst + r * KROW + lane * 4);
            const float* g = src + (size_t)r * DIM;
            // GLOBAL_LOAD_ASYNC_TO_LDS_B128: per-lane 16B, tracked by ASYNCcnt
            asm volatile("global_load_async_to_lds_b128 %0, %1, off"
                         :: "v"(lds_off), "v"(g)
                         : "memory");
        }
    };

    // ---------- load Q tile as WMMA A-fragments (16x4 f32 layout) ----------
    // A layout: v0 = {K=0 | K=2}, v1 = {K=1 | K=3}; M = lane%16.
    v2f qfrag[32];
    {
        const float* qbase = h_dec + ((size_t)(q0 + lm) * NBATCH + b) * DIM + d0 + half * 2;
#pragma unroll
        for (int s = 0; s < 32; ++s) {
            const float* p = qbase + 4 * s;
            v2f q = {p[0], p[1]};
            qfrag[s] = q;
        }
    }

    v8f oacc[8];
#pragma unroll
    for (int t = 0; t < 8; ++t) oacc[t] = (v8f){0, 0, 0, 0, 0, 0, 0, 0};
    float mrow[8], lrow[8];
#pragma unroll
    for (int i = 0; i < 8; ++i) { mrow[i] = -3.0e38f; lrow[i] = 0.0f; }

    issue_kblock(0);

    for (int kb = 0; kb < NKB; ++kb) {
        // Refill the other buffer (its readers finished last iteration; drain DS
        // first since async LDS writes are unordered vs this wave's DS reads),
        // then wait for the current block's 16 async loads (in-order completion).
        if (kb + 1 < NKB) {
            asm volatile("s_wait_dscnt 0" ::: "memory");
            issue_kblock(kb + 1);
            asm volatile("s_wait_asynccnt 16" ::: "memory");
        } else {
            asm volatile("s_wait_asynccnt 0" ::: "memory");
        }
        const float* kcur = smem + (((kb & 1) * WAVES + wave) * (NKEYS * KROW));
        float* sred = sred0 + (kb & 1) * SREDSZ;

        // L2 prefetch two blocks ahead (global_prefetch_b8)
        if (kb + 2 < NKB) {
            __builtin_prefetch(encb + (size_t)((kb + 2) * NKEYS + lm) * DIM + d0 + half * 64, 0, 0);
        }

        // ---------- partial scores S_w[16x16] = Q_chunk * K_chunk^T ----------
        // B layout (4x16): v0 = rows {0|2}, v1 = rows {1|3}; N = lane%16 = key.
        // Two independent accumulator chains; B fragments preloaded in groups of 8.
        v8f sacc0 = (v8f){0, 0, 0, 0, 0, 0, 0, 0};
        v8f sacc1 = (v8f){0, 0, 0, 0, 0, 0, 0, 0};
#pragma unroll
        for (int g = 0; g < 4; ++g) {
            v2f bf[8];
#pragma unroll
            for (int j = 0; j < 8; ++j) {
                const float* kp = kcur + lm * KROW + 4 * (8 * g + j) + half * 2;
                v2f t = {kp[0], kp[1]};
                bf[j] = t;
            }
#pragma unroll
            for (int j = 0; j < 8; ++j) {
                if ((j & 1) == 0)
                    sacc0 = __builtin_amdgcn_wmma_f32_16x16x4_f32(
                        false, qfrag[8 * g + j], false, bf[j], (short)0, sacc0, false, false);
                else
                    sacc1 = __builtin_amdgcn_wmma_f32_16x16x4_f32(
                        false, qfrag[8 * g + j], false, bf[j], (short)0, sacc1, false, false);
            }
        }

        // ---------- cross-wave reduction of scores over the 8 D-chunks ----------
        // layout: sred[i][wave][lane] -> the 8 partials per lane are 128B apart
#pragma unroll
        for (int i = 0; i < 8; ++i)
            sred[i * (WAVES * 32) + wave * 32 + lane] = sacc0[i] + sacc1[i];
        __syncthreads();   // single barrier per block (sred is parity double-buffered)
        float sfull[8];
#pragma unroll
        for (int i = 0; i < 8; ++i) {
            const float* sp = sred + i * (WAVES * 32) + lane;
            float p0 = sp[0 * 32] + sp[1 * 32];
            float p1 = sp[2 * 32] + sp[3 * 32];
            float p2 = sp[4 * 32] + sp[5 * 32];
            float p3 = sp[6 * 32] + sp[7 * 32];
            sfull[i] = (p0 + p1) + (p2 + p3);
        }

        // ---------- online softmax (C layout: row = i + 8*half, col/key = lane%16) ----------
        float pval[8], alpha[8];
        int needscale = 0;
#pragma unroll
        for (int i = 0; i < 8; ++i) {
            float rmax = sfull[i];
            rmax = fmaxf(rmax, __shfl_xor(rmax, 1));
            rmax = fmaxf(rmax, __shfl_xor(rmax, 2));
            rmax = fmaxf(rmax, __shfl_xor(rmax, 4));
            rmax = fmaxf(rmax, __shfl_xor(rmax, 8));   // max over 16 keys (stays in half)
            const float mnew = fmaxf(mrow[i], rmax);
            const float a  = __expf(mrow[i] - mnew);   // exactly 1.0f when max unchanged
            const float pv = __expf(sfull[i] - mnew);
            float rsum = pv;
            rsum += __shfl_xor(rsum, 1);
            rsum += __shfl_xor(rsum, 2);
            rsum += __shfl_xor(rsum, 4);
            rsum += __shfl_xor(rsum, 8);
            lrow[i]  = lrow[i] * a + rsum;
            mrow[i]  = mnew;
            alpha[i] = a;
            pval[i]  = pv;
            needscale |= (a != 1.0f);
        }
        // steady state: row max rarely changes -> skip the 64-mul rescale
        if (__any(needscale)) {
#pragma unroll
            for (int t = 0; t < 8; ++t)
#pragma unroll
                for (int i = 0; i < 8; ++i) oacc[t][i] *= alpha[i];
        }

        // ---------- redistribute P (C layout) -> A layout via wave-private LDS ----------
#pragma unroll
        for (int i = 0; i < 8; ++i) pbuf[(i + 8 * half) * 17 + lm] = pval[i];
        // same-wave DS ops are in-order; compiler inserts s_wait_dscnt for the RAW

        // ---------- O_chunk += P * V_chunk ----------
        // A = P[16q x 16key]; B[key_sub, d_col] = V[e0+key][d0+16t+lm]
#pragma unroll
        for (int s = 0; s < 4; ++s) {
            const float* pp = pbuf + lm * 17 + 4 * s + half * 2;
            v2f af = {pp[0], pp[1]};
            const int krow = 4 * s + half * 2;
            v2f bf[8];
#pragma unroll
            for (int t = 0; t < 8; ++t) {
                bf[t][0] = kcur[(size_t)krow * KROW + 16 * t + lm];
                bf[t][1] = kcur[(size_t)(krow + 1) * KROW + 16 * t + lm];
            }
#pragma unroll
            for (int t = 0; t < 8; ++t)
                oacc[t] = __builtin_amdgcn_wmma_f32_16x16x4_f32(
                    false, af, false, bf[t], (short)0, oacc[t], false, false);
        }
    }

    // ---------- epilogue: divide by row sums, store ----------
    float inv[8];
#pragma unroll
    for (int i = 0; i < 8; ++i) inv[i] = 1.0f / lrow[i];

    float* ob = out + ((size_t)b * TDEC + q0) * DIM + d0;
#pragma unroll
    for (int t = 0; t < 8; ++t)
#pragma unroll
        for (int i = 0; i < 8; ++i)
            ob[(size_t)(i + 8 * half) * DIM + 16 * t + lm] = oacc[t][i] * inv[i];
}

extern "C" void kernel_launch(void* const* d_in, const int* in_sizes, int n_in,
                              void* d_out, int out_size, void* d_ws, size_t ws_size,
                              hipStream_t stream) {
    (void)in_sizes; (void)n_in; (void)out_size; (void)d_ws; (void)ws_size;
    const float* h_enc = (const float*)d_in[0];
    const float* h_dec = (const float*)d_in[1];
    float* out = (float*)d_out;

    const size_t lds_bytes =
        (size_t)(2 * WAVES * NKEYS * KROW   // double-buffered K/V
                 + 2 * SREDSZ               // double-buffered score reduce
                 + WAVES * 16 * 17)         // P scratch
        * sizeof(float);                    // ~160 KB -> 2 workgroups per WGP

    // allow >64KB dynamic LDS (CDNA5 WGP has 320KB)
    hipFuncSetAttribute((const void*)xattn_flash_f32,
                        hipFuncAttributeMaxDynamicSharedMemorySize, (int)lds_bytes);

    dim3 grid(TDEC / 16, NBATCH);   // 32 query tiles x 16 batches
    dim3 block(256);                // 8 wave32
    hipLaunchKernelGGL(xattn_flash_f32, grid, block, lds_bytes, stream, h_enc, h_dec, out);
}